// Net_90855738179662
// MI455X (gfx1250) — compile-verified
//
#include <hip/hip_runtime.h>
#include <math.h>

#define N_NODES 50000
#define N_EDGES 800000
#define N_TOT   (N_EDGES + N_NODES)
#define N_GRAPH 512
#define BN_EPS  1e-5f

typedef __attribute__((ext_vector_type(2))) float v2f;
typedef __attribute__((ext_vector_type(8))) float v8f;

__device__ __forceinline__ float lrelu(float x, float s) { return x > 0.f ? x : s * x; }

// ---------------------------------------------------------------------------
// C[M,F] = A[M,K] @ W[K,F]   (M = N_NODES mult of 16; K mult of 4; F mult of 16)
// One wave per 16x16 tile, V_WMMA_F32_16X16X4_F32 along K (exact fp32).
// Fragment layout per CDNA5 ISA 7.12.2:
//   A 16x4 : lane(m, half) holds {A[m][k+2*half], A[m][k+2*half+1]}
//   B 4x16 : lane(n, half) holds {W[k+2*half][n], W[k+2*half+1][n]}
//   D 16x16: VGPR r -> row r + 8*half, col = lane&15
// ---------------------------------------------------------------------------
__global__ void gemm_wmma(const float* __restrict__ A, const float* __restrict__ W,
                          float* __restrict__ C, int K, int F) {
    const int lane   = threadIdx.x & 31;
    const int wave   = blockIdx.x * (blockDim.x >> 5) + (threadIdx.x >> 5);
    const int tilesN = F >> 4;
    const int tileM  = wave / tilesN;
    const int tileN  = wave % tilesN;
    if (tileM >= (N_NODES >> 4)) return;
    const int half = lane >> 4;
    const int lm   = lane & 15;
    const int row  = (tileM << 4) + lm;   // A row
    const int col  = (tileN << 4) + lm;   // B/C col
    const float* __restrict__ arow = A + (size_t)row * K;

    v8f acc = {};
    for (int k = 0; k < K; k += 4) {
        const int kk = k + 2 * half;
        v2f a, b;
        a.x = arow[kk];
        a.y = arow[kk + 1];
        b.x = W[(size_t)kk * F + col];
        b.y = W[(size_t)(kk + 1) * F + col];
        acc = __builtin_amdgcn_wmma_f32_16x16x4_f32(false, a, false, b,
                                                    (short)0, acc, false, false);
    }
    const int mbase = (tileM << 4) + (half << 3);
#pragma unroll
    for (int r = 0; r < 8; ++r)
        C[(size_t)(mbase + r) * F + col] = acc[r];
}

// a_src[i] = h[i]·a_s, a_dst[i] = h[i]·a_d
__global__ void node_dots(const float* __restrict__ H, const float* __restrict__ a_s,
                          const float* __restrict__ a_d, float* __restrict__ asrc,
                          float* __restrict__ adst, int F) {
    int i = blockIdx.x * blockDim.x + threadIdx.x;
    if (i >= N_NODES) return;
    const float* h = H + (size_t)i * F;
    float s = 0.f, d = 0.f;
    for (int f = 0; f < F; ++f) { float v = h[f]; s += v * a_s[f]; d += v * a_d[f]; }
    asrc[i] = s; adst[i] = d;
}

__global__ void fill_zero_f(float* __restrict__ p, int n) {
    int i = blockIdx.x * blockDim.x + threadIdx.x;
    if (i < n) p[i] = 0.f;
}
__global__ void fill_zero_i(int* __restrict__ p, int n) {
    int i = blockIdx.x * blockDim.x + threadIdx.x;
    if (i < n) p[i] = 0;
}
__global__ void copy_i(const int* __restrict__ a, int* __restrict__ b, int n) {
    int i = blockIdx.x * blockDim.x + threadIdx.x;
    if (i < n) b[i] = a[i];
}

// ---------------- CSR build (once per call) ----------------
__global__ void edge_hist(const int* __restrict__ dst, int* __restrict__ cnt) {
    int e = blockIdx.x * blockDim.x + threadIdx.x;
    if (e >= N_TOT) return;
    int d = (e < N_EDGES) ? dst[e] : (e - N_EDGES);
    atomicAdd(&cnt[d], 1);
}

// single-block chunked Hillis-Steele exclusive scan: off[0..N] from cnt[0..N-1]
__global__ void excl_scan(const int* __restrict__ cnt, int* __restrict__ off) {
    __shared__ int sh[1024];
    __shared__ int carry;
    const int t = threadIdx.x;
    if (t == 0) carry = 0;
    __syncthreads();
    for (int base = 0; base < N_NODES; base += 1024) {
        int i = base + t;
        int v = (i < N_NODES) ? cnt[i] : 0;
        sh[t] = v;
        __syncthreads();
        for (int o = 1; o < 1024; o <<= 1) {
            int add = (t >= o) ? sh[t - o] : 0;
            __syncthreads();
            sh[t] += add;
            __syncthreads();
        }
        if (i < N_NODES) off[i] = carry + sh[t] - v;   // exclusive
        __syncthreads();
        if (t == 1023) carry += sh[1023];
        __syncthreads();
    }
    if (t == 0) off[N_NODES] = carry;
}

__global__ void edge_bucket(const int* __restrict__ dst, int* __restrict__ cursor,
                            int* __restrict__ perm) {
    int e = blockIdx.x * blockDim.x + threadIdx.x;
    if (e >= N_TOT) return;
    int d = (e < N_EDGES) ? dst[e] : (e - N_EDGES);
    int pos = atomicAdd(&cursor[d], 1);
    perm[pos] = e;
}

// ---------------- per-layer attention, atomic-free ----------------
// one thread per dst node: pass 1 caches logits + running max; pass 2 sums exp
__global__ void attn_stats(const int* __restrict__ srcA, const int* __restrict__ off,
                           const int* __restrict__ perm, const float* __restrict__ asrc,
                           const float* __restrict__ adst, float* __restrict__ wslot,
                           float* __restrict__ mbuf, float* __restrict__ dbuf) {
    int d = blockIdx.x * blockDim.x + threadIdx.x;
    if (d >= N_NODES) return;
    const int b = off[d], e2 = off[d + 1];
    const float ad = adst[d];
    float m = -INFINITY;
    for (int j = b; j < e2; ++j) {
        int e = perm[j];
        int s = (e < N_EDGES) ? srcA[e] : (e - N_EDGES);
        float v = lrelu(asrc[s] + ad, 0.2f);
        wslot[j] = v;
        m = fmaxf(m, v);
    }
    float den = 0.f;
    for (int j = b; j < e2; ++j) den += expf(wslot[j] - m);
    mbuf[d] = m;
    dbuf[d] = den;
}

// one wave per dst node, lanes striped over features: out[d] = sum alpha*h[src],
// fused + bias + leaky_relu(0.01). No atomics; each row written exactly once.
__global__ void aggregate(const int* __restrict__ srcA, const int* __restrict__ off,
                          const int* __restrict__ perm, const float* __restrict__ wslot,
                          const float* __restrict__ mbuf, const float* __restrict__ dbuf,
                          const float* __restrict__ H, const float* __restrict__ bias,
                          float* __restrict__ out, int F) {
    const int node = blockIdx.x * (blockDim.x >> 5) + (threadIdx.x >> 5);
    if (node >= N_NODES) return;
    const int lane = threadIdx.x & 31;
    const int b = off[node], e2 = off[node + 1];
    const float m = mbuf[node];
    const float invden = 1.f / dbuf[node];
    const int C = F >> 5;                 // 1, 2 or 4
    float acc[4] = {0.f, 0.f, 0.f, 0.f};
    for (int j = b; j < e2; ++j) {
        int e = perm[j];
        int s = (e < N_EDGES) ? srcA[e] : (e - N_EDGES);
        float alpha = expf(wslot[j] - m) * invden;
        const float* __restrict__ hs = H + (size_t)s * F;
        for (int c = 0; c < C; ++c) acc[c] += alpha * hs[(c << 5) + lane];
    }
    float* o = out + (size_t)node * F;
    for (int c = 0; c < C; ++c) {
        int f = (c << 5) + lane;
        o[f] = lrelu(acc[c] + bias[f], 0.01f);
    }
}

// one block per feature: biased mean/var -> mean[f], inv[f] = rsqrt(var+eps)
__global__ void bn_stats(const float* __restrict__ acc, float* __restrict__ mean,
                         float* __restrict__ inv, int F) {
    const int f = blockIdx.x, t = threadIdx.x;
    float s = 0.f, s2 = 0.f;
    for (int i = t; i < N_NODES; i += blockDim.x) {
        float v = acc[(size_t)i * F + f];
        s += v; s2 += v * v;
    }
    __shared__ float sh[256], sh2[256];
    sh[t] = s; sh2[t] = s2; __syncthreads();
    for (int o = 128; o > 0; o >>= 1) {
        if (t < o) { sh[t] += sh[t + o]; sh2[t] += sh2[t + o]; }
        __syncthreads();
    }
    if (t == 0) {
        float mu  = sh[0] / (float)N_NODES;
        float var = sh2[0] / (float)N_NODES - mu * mu;
        mean[f] = mu;
        inv[f]  = rsqrtf(var + BN_EPS);
    }
}

__global__ void bn_apply(float* __restrict__ acc, const float* __restrict__ mean,
                         const float* __restrict__ inv, const float* __restrict__ g,
                         const float* __restrict__ be, int F) {
    int tid = blockIdx.x * blockDim.x + threadIdx.x;
    if (tid >= N_NODES * F) return;
    int f = tid % F;
    acc[tid] = (acc[tid] - mean[f]) * inv[f] * g[f] + be[f];
}

__global__ void pool_add(const float* __restrict__ h, const int* __restrict__ batch,
                         float* __restrict__ pooled) {
    int tid = blockIdx.x * blockDim.x + threadIdx.x;
    if (tid >= N_NODES * 64) return;
    int i = tid >> 6, f = tid & 63;
    atomicAdd(&pooled[((size_t)batch[i] << 6) + f], h[tid]);
}

// one wave per graph: 64->32 lrelu -> 32->8 sigmoid
__global__ void head(const float* __restrict__ pooled, const float* __restrict__ Wf,
                     const float* __restrict__ bf, const float* __restrict__ Wc,
                     const float* __restrict__ bc, float* __restrict__ out) {
    __shared__ float z[32];
    const int g = blockIdx.x, t = threadIdx.x;
    const float* p = pooled + ((size_t)g << 6);
    float s = bf[t];
    for (int k = 0; k < 64; ++k) s += p[k] * Wf[k * 32 + t];
    z[t] = lrelu(s, 0.01f);
    __syncthreads();
    if (t < 8) {
        float o = bc[t];
        for (int k = 0; k < 32; ++k) o += z[k] * Wc[k * 8 + t];
        out[(size_t)g * 8 + t] = 1.f / (1.f + expf(-o));
    }
}

// ---------------------------------------------------------------------------
extern "C" void kernel_launch(void* const* d_in, const int* in_sizes, int n_in,
                              void* d_out, int out_size, void* d_ws, size_t ws_size,
                              hipStream_t stream) {
    (void)in_sizes; (void)n_in; (void)out_size; (void)ws_size;

    const float* x   = (const float*)d_in[0];
    const int*   ei  = (const int*)d_in[1];       // [2,E] flat
    const int*   bat = (const int*)d_in[2];
    const int* srcA = ei;
    const int* dstA = ei + N_EDGES;

    // ---- workspace layout ----
    const size_t NB = (size_t)N_NODES * 128;
    float* w0 = (float*)d_ws;
    float* bufs[3]  = { w0, w0 + NB, w0 + 2 * NB };
    float* asrc     = w0 + 3 * NB;
    float* adst     = asrc + N_NODES;
    float* mbuf     = adst + N_NODES;
    float* dbuf     = mbuf + N_NODES;
    float* wslot    = dbuf + N_NODES;                 // N_TOT logits (CSR order)
    float* pooled   = wslot + N_TOT;
    float* meanv    = pooled + (size_t)N_GRAPH * 64;
    float* invv     = meanv + 128;
    int*   cnt      = (int*)(invv + 128);
    int*   off      = cnt + N_NODES;                  // N_NODES+1
    int*   cursor   = off + N_NODES + 1;
    int*   perm     = cursor + N_NODES;               // N_TOT

    const int T = 256;

    // ---- CSR build by dst (self-loops are edge ids E..E+N-1) ----
    fill_zero_i<<<(N_NODES + T - 1) / T, T, 0, stream>>>(cnt, N_NODES);
    edge_hist<<<(N_TOT + T - 1) / T, T, 0, stream>>>(dstA, cnt);
    excl_scan<<<1, 1024, 0, stream>>>(cnt, off);
    copy_i<<<(N_NODES + T - 1) / T, T, 0, stream>>>(off, cursor, N_NODES);
    edge_bucket<<<(N_TOT + T - 1) / T, T, 0, stream>>>(dstA, cursor, perm);

    const int dins[4]   = {128, 32, 64, 128};
    const int douts[4]  = {32, 64, 128, 64};
    const int tmpIdx[4] = {0, 2, 1, 0};
    const int accIdx[4] = {1, 0, 2, 1};

    const float* cur = x;
    for (int l = 0; l < 4; ++l) {
        const int K = dins[l], F = douts[l];
        float* htmp = bufs[tmpIdx[l]];
        float* acc  = bufs[accIdx[l]];
        const float* Wl  = (const float*)d_in[3 + 6 * l + 0];
        const float* asl = (const float*)d_in[3 + 6 * l + 1];
        const float* adl = (const float*)d_in[3 + 6 * l + 2];
        const float* bl  = (const float*)d_in[3 + 6 * l + 3];
        const float* gl  = (const float*)d_in[3 + 6 * l + 4];
        const float* bel = (const float*)d_in[3 + 6 * l + 5];

        const int waves = (N_NODES / 16) * (F / 16);
        gemm_wmma<<<(waves + 7) / 8, T, 0, stream>>>(cur, Wl, htmp, K, F);
        node_dots<<<(N_NODES + T - 1) / T, T, 0, stream>>>(htmp, asl, adl, asrc, adst, F);

        attn_stats<<<(N_NODES + T - 1) / T, T, 0, stream>>>(srcA, off, perm, asrc, adst,
                                                            wslot, mbuf, dbuf);
        const int aggBlocks = (N_NODES * 32 + T - 1) / T;   // one wave per node
        aggregate<<<aggBlocks, T, 0, stream>>>(srcA, off, perm, wslot, mbuf, dbuf,
                                               htmp, bl, acc, F);

        bn_stats<<<F, 256, 0, stream>>>(acc, meanv, invv, F);
        bn_apply<<<(N_NODES * F + T - 1) / T, T, 0, stream>>>(acc, meanv, invv, gl, bel, F);

        cur = acc;
    }

    fill_zero_f<<<(N_GRAPH * 64 + T - 1) / T, T, 0, stream>>>(pooled, N_GRAPH * 64);
    pool_add<<<(N_NODES * 64 + T - 1) / T, T, 0, stream>>>(cur, bat, pooled);

    const float* Wf = (const float*)d_in[27];
    const float* bf = (const float*)d_in[28];
    const float* Wc = (const float*)d_in[29];
    const float* bc = (const float*)d_in[30];
    head<<<N_GRAPH, 32, 0, stream>>>(pooled, Wf, bf, Wc, bc, (float*)d_out);
}